// StrangeLoopEncoder_89781996356352
// MI455X (gfx1250) — compile-verified
//
#include <hip/hip_runtime.h>
#include <hip/hip_bf16.h>

// ---------------------------------------------------------------------------
// StrangeLoopEncoder collapsed to:
//   out[(b,t),:] = x[(b,t),:] @ (Z @ W0^T + 0.1^13 * A^13) + c*(b0 + godel[t])
//   A = S + 0.1*S*S^T,  Z = sum_{d=0..12} (0.1*A)^d,  c = (1-0.1^13)/0.9
// All GEMMs run on the fp32 matrix pipe: V_WMMA_F32_16X16X4_F32 (wave32).
// ---------------------------------------------------------------------------

typedef __attribute__((ext_vector_type(2))) float v2f;
typedef __attribute__((ext_vector_type(4))) float v4f;
typedef __attribute__((ext_vector_type(8))) float v8f;

#define GK 1024   // inner (K) dimension, fixed
#define GN 1024   // output columns, fixed
#define BM 64     // block tile M
#define BN 64     // block tile N
#define KC 32     // K chunk staged in LDS
#define PAD 8     // LDS row pad: 2 rows = 144 floats = +16 banks -> halves disjoint

// General fp32-WMMA GEMM:  Cout = alpha*(A x Beff) [+ beta*Cadd]
//                                [+ bias_scale*(bias_n[n] + bias_rows[m%Tmod][n])]
// and optionally  Zacc += zcoef * Cout   (power-series accumulation).
// Beff[k][n] = transB ? Bm[n][k] : Bm[k][n].  All matrices row-major, ld = 1024.
__global__ __launch_bounds__(256) void gemm_wmma_f32(
    const float* __restrict__ Am, const float* __restrict__ Bm, int transB,
    float alpha, const float* __restrict__ Cadd, float beta,
    float* __restrict__ Cout, float* __restrict__ Zacc, float zcoef,
    const float* __restrict__ bias_n, const float* __restrict__ bias_rows,
    float bias_scale, int Tmod)
{
    __shared__ float As[KC][BM + PAD];   // A staged transposed: As[k][m]
    __shared__ float Bs[KC][BN + PAD];   // Bs[k][n]

    const int tid   = threadIdx.x;
    const int lane  = tid & 31;
    const int w     = tid >> 5;          // 8 waves
    const int waveM = w >> 1;            // 0..3 -> 16-row slab
    const int waveN = w & 1;             // 0..1 -> 32-col slab
    const int hl    = lane >> 4;         // lane half (K split per ISA layout)
    const int l     = lane & 15;

    const int rowBase = blockIdx.y * BM;
    const int colBase = blockIdx.x * BN;
    const int Arow0   = waveM * 16;
    const int Bcol0   = waveN * 32;

    // loader indices
    const int ar  = tid >> 2;            // 0..63 : A tile row
    const int aq  = tid & 3;             // 0..3  : 8-float column group
    const int bkr = tid >> 4;            // 0..15 : B (no-trans) k row
    const int bc4 = tid & 15;            //         B (no-trans) col quad

    v8f acc0 = {};
    v8f acc1 = {};

    for (int k0 = 0; k0 < GK; k0 += KC) {
        // ---- stage A (64 x 32), transposed into LDS ----
        {
            const float* src = Am + (size_t)(rowBase + ar) * GK + k0 + aq * 8;
            v4f p0 = *(const v4f*)(src);
            v4f p1 = *(const v4f*)(src + 4);
#pragma unroll
            for (int j = 0; j < 4; ++j) As[aq * 8 + j][ar] = p0[j];
#pragma unroll
            for (int j = 0; j < 4; ++j) As[aq * 8 + 4 + j][ar] = p1[j];
        }
        // ---- stage B (32 x 64) ----
        if (transB == 0) {
            const float* s0 = Bm + (size_t)(k0 + bkr) * GN + colBase + bc4 * 4;
            const float* s1 = Bm + (size_t)(k0 + bkr + 16) * GN + colBase + bc4 * 4;
            *(v4f*)&Bs[bkr][bc4 * 4]      = *(const v4f*)s0;
            *(v4f*)&Bs[bkr + 16][bc4 * 4] = *(const v4f*)s1;
        } else {
            const float* src = Bm + (size_t)(colBase + ar) * GK + k0 + aq * 8;
            v4f p0 = *(const v4f*)(src);
            v4f p1 = *(const v4f*)(src + 4);
#pragma unroll
            for (int j = 0; j < 4; ++j) Bs[aq * 8 + j][ar] = p0[j];
#pragma unroll
            for (int j = 0; j < 4; ++j) Bs[aq * 8 + 4 + j][ar] = p1[j];
        }
        __syncthreads();

        // ---- 8 x (16x16x4) WMMA steps per chunk, 2 N-subtiles per wave ----
#pragma unroll
        for (int kk = 0; kk < KC; kk += 4) {
            const int ko = kk + (hl << 1);   // lanes 0-15 -> K{kk,kk+1}; 16-31 -> K{kk+2,kk+3}
            v2f a, b0v, b1v;
            a[0]   = As[ko][Arow0 + l];
            a[1]   = As[ko + 1][Arow0 + l];
            b0v[0] = Bs[ko][Bcol0 + l];
            b0v[1] = Bs[ko + 1][Bcol0 + l];
            b1v[0] = Bs[ko][Bcol0 + 16 + l];
            b1v[1] = Bs[ko + 1][Bcol0 + 16 + l];
            acc0 = __builtin_amdgcn_wmma_f32_16x16x4_f32(false, a, false, b0v,
                                                         (short)0, acc0, false, false);
            acc1 = __builtin_amdgcn_wmma_f32_16x16x4_f32(false, a, false, b1v,
                                                         (short)0, acc1, false, false);
        }
        __syncthreads();
    }

    // ---- epilogue (C/D layout: VGPR v -> M = v + 8*half, lane -> N) ----
#pragma unroll
    for (int nt = 0; nt < 2; ++nt) {
        v8f acc = nt ? acc1 : acc0;
        const int ng = colBase + Bcol0 + nt * 16 + l;
#pragma unroll
        for (int v = 0; v < 8; ++v) {
            const int mg = rowBase + Arow0 + v + hl * 8;
            const size_t idx = (size_t)mg * GN + ng;
            float val = alpha * acc[v];
            if (Cadd)      val += beta * Cadd[idx];
            if (bias_n)    val += bias_scale * bias_n[ng];
            if (bias_rows) val += bias_scale * bias_rows[(size_t)(mg % Tmod) * GN + ng];
            Cout[idx] = val;
            if (Zacc) Zacc[idx] += zcoef * val;
        }
    }
}

// Z = I + 0.1*A (d=0,1 terms of the geometric series), P = A  (fresh every call)
__global__ void init_zp(const float* __restrict__ A, float* __restrict__ Z,
                        float* __restrict__ P)
{
    const int i = blockIdx.x * blockDim.x + threadIdx.x;   // 0 .. 1024*1024-1
    const float a = A[i];
    const int r = i >> 10, c = i & 1023;
    Z[i] = 0.1f * a + ((r == c) ? 1.0f : 0.0f);
    P[i] = a;
}

static inline void launch_gemm(const float* Am, const float* Bm, int transB,
                               float alpha, const float* Cadd, float beta,
                               float* Cout, float* Zacc, float zcoef,
                               const float* bias_n, const float* bias_rows,
                               float bias_scale, int Tmod, int M,
                               hipStream_t stream)
{
    dim3 grid(GN / BN, M / BM);
    gemm_wmma_f32<<<grid, 256, 0, stream>>>(Am, Bm, transB, alpha, Cadd, beta,
                                            Cout, Zacc, zcoef, bias_n, bias_rows,
                                            bias_scale, Tmod);
}

extern "C" void kernel_launch(void* const* d_in, const int* in_sizes, int n_in,
                              void* d_out, int out_size, void* d_ws, size_t ws_size,
                              hipStream_t stream)
{
    // setup_inputs order: x, spiral_weights(10), spiral_biases(10), godel_table, self_ref
    const float *x, *W0, *b0, *godel, *S;
    if (n_in >= 23) {
        x     = (const float*)d_in[0];
        W0    = (const float*)d_in[1];    // layer-0 weight is all we need
        b0    = (const float*)d_in[11];   // layer-0 bias
        godel = (const float*)d_in[21];
        S     = (const float*)d_in[22];
    } else {                               // tuples packed flat: layer-0 is at the head
        x     = (const float*)d_in[0];
        W0    = (const float*)d_in[1];
        b0    = (const float*)d_in[2];
        godel = (const float*)d_in[3];
        S     = (const float*)d_in[4];
    }

    float* ws = (float*)d_ws;
    const size_t MM = (size_t)1024 * 1024;
    float* A  = ws;            // A = S + 0.1*S*S^T
    float* P0 = ws + MM;       // power ping
    float* P1 = ws + 2 * MM;   // power pong
    float* Z  = ws + 3 * MM;   // geometric series accumulator
    float* Mt = ws + 4 * MM;   // fused matrix  Z*W0^T + 1e-13*A^13

    // 1) A = 0.1*(S*S^T) + S
    launch_gemm(S, S, /*transB=*/1, 0.1f, /*Cadd=*/S, 1.0f, A,
                nullptr, 0.f, nullptr, nullptr, 0.f, 1, 1024, stream);

    // 2) Z = I + 0.1*A ;  P0 = A
    init_zp<<<dim3(4096), dim3(256), 0, stream>>>(A, Z, P0);

    // 3) powers A^2 .. A^13 ; accumulate Z += (0.1^d)*A^d for d<=12
    float* Pprev = P0;
    float* Pcur  = P1;
    double coef  = 0.1;
    for (int d = 2; d <= 13; ++d) {
        coef *= 0.1;                                   // 0.1^d
        float* zptr = (d <= 12) ? Z : nullptr;
        launch_gemm(Pprev, A, /*transB=*/0, 1.0f, nullptr, 0.f, Pcur,
                    zptr, (float)coef, nullptr, nullptr, 0.f, 1, 1024, stream);
        float* t = Pprev; Pprev = Pcur; Pcur = t;
    }
    // Pprev now holds A^13

    // 4) Mt = Z * W0^T + 1e-13 * A^13
    launch_gemm(Z, W0, /*transB=*/1, 1.0f, /*Cadd=*/Pprev, 1e-13f, Mt,
                nullptr, 0.f, nullptr, nullptr, 0.f, 1, 1024, stream);

    // 5) out = x * Mt + c*(b0[n] + godel[t][n]),  t = row % 2048
    const float c = (float)((1.0 - 1e-13) / 0.9);
    launch_gemm(x, Mt, /*transB=*/0, 1.0f, nullptr, 0.f, (float*)d_out,
                nullptr, 0.f, /*bias_n=*/b0, /*bias_rows=*/godel, c,
                /*Tmod=*/2048, /*M=*/16384, stream);
}